// BaselineGNN_8512625180983
// MI455X (gfx1250) — compile-verified
//
#include <hip/hip_runtime.h>
#include <math.h>

// Problem constants (match reference)
#define NN 50000   // nodes
#define NE 800000  // edges
#define NG 500     // graphs
#define HD 64      // hidden

typedef __attribute__((ext_vector_type(2))) float v2f;
typedef __attribute__((ext_vector_type(8))) float v8f;

__device__ __forceinline__ float silu_f(float x) { return x / (1.0f + __expf(-x)); }

// ---------------------------------------------------------------------------
// Zero-fill kernel (agg buffer + pool accumulators must start at 0)
// ---------------------------------------------------------------------------
__global__ void zero_kernel(float* __restrict__ p, long n) {
    long i = (long)blockIdx.x * blockDim.x + threadIdx.x;
    if (i < n) p[i] = 0.0f;
}

// ---------------------------------------------------------------------------
// Node embedding: h = SiLU(LN(x @ ne_w + ne_b))   [N,16] -> [N,64]
// One wave (32 lanes) per row; each lane owns cols {lane, lane+32}.
// ---------------------------------------------------------------------------
__global__ void __launch_bounds__(128) node_embed_kernel(
    const float* __restrict__ x, const float* __restrict__ w,
    const float* __restrict__ b, const float* __restrict__ g,
    const float* __restrict__ be, float* __restrict__ out, int nrows) {
    int row  = (int)((blockIdx.x * blockDim.x + threadIdx.x) >> 5);
    int lane = threadIdx.x & 31;
    if (row >= nrows) return;
    const float* xr = x + (long)row * 16;
    float a0 = b[lane], a1 = b[lane + 32];
#pragma unroll
    for (int k = 0; k < 16; ++k) {
        float xv = xr[k];
        a0 += xv * w[k * 64 + lane];
        a1 += xv * w[k * 64 + lane + 32];
    }
    float s = a0 + a1, ss = a0 * a0 + a1 * a1;
#pragma unroll
    for (int xm = 1; xm < 32; xm <<= 1) { s += __shfl_xor(s, xm); ss += __shfl_xor(ss, xm); }
    float mean = s * (1.0f / 64.0f);
    float rstd = rsqrtf(ss * (1.0f / 64.0f) - mean * mean + 1e-5f);
    out[(long)row * 64 + lane]      = silu_f((a0 - mean) * rstd * g[lane]      + be[lane]);
    out[(long)row * 64 + lane + 32] = silu_f((a1 - mean) * rstd * g[lane + 32] + be[lane + 32]);
}

// ---------------------------------------------------------------------------
// Fused message kernel (one wave = one 16-edge tile):
//   e     = SiLU(LN(edge_attr @ ee_w + ee_b))            (recomputed -> LDS)
//   inp   = [h[dst] | h[src] | e]                         [16,192]
//   m     = LN(SiLU(inp @ W + mb))                        via WMMA f32 16x16x4
//   agg[dst] += m                                         (f32 global atomics)
// Grid is EXACT (NE/16 tiles, 4 waves/block) so __syncthreads is safe.
// ---------------------------------------------------------------------------
__global__ void __launch_bounds__(128) msg_kernel(
    const float* __restrict__ h, const float* __restrict__ edge_attr,
    const float* __restrict__ ee_w, const float* __restrict__ ee_b,
    const float* __restrict__ ee_g, const float* __restrict__ ee_be,
    const float* __restrict__ W,  // [192,64] slice for this layer
    const float* __restrict__ mb, const float* __restrict__ mg,
    const float* __restrict__ mbe,
    const int* __restrict__ srcv, const int* __restrict__ dstv,
    float* __restrict__ agg) {
    __shared__ float lds_e[4][16][68];  // pad 68: conflict-free strided A reads
    int wib  = threadIdx.x >> 5;
    int tile = blockIdx.x * 4 + wib;
    int lane = threadIdx.x & 31;
    int m    = lane & 15;
    int half = lane >> 4;
    int e0   = tile * 16;

    // ---- edge embedding for this tile's 16 edges -> LDS ----
    float (*le)[68] = lds_e[wib];
    for (int r = 0; r < 16; ++r) {
        const float* ar = edge_attr + (long)(e0 + r) * 8;
        float a0 = ee_b[lane], a1 = ee_b[lane + 32];
#pragma unroll
        for (int k = 0; k < 8; ++k) {
            float xv = ar[k];
            a0 += xv * ee_w[k * 64 + lane];
            a1 += xv * ee_w[k * 64 + lane + 32];
        }
        float s = a0 + a1, ss = a0 * a0 + a1 * a1;
#pragma unroll
        for (int xm = 1; xm < 32; xm <<= 1) { s += __shfl_xor(s, xm); ss += __shfl_xor(ss, xm); }
        float mean = s * (1.0f / 64.0f);
        float rstd = rsqrtf(ss * (1.0f / 64.0f) - mean * mean + 1e-5f);
        le[r][lane]      = silu_f((a0 - mean) * rstd * ee_g[lane]      + ee_be[lane]);
        le[r][lane + 32] = silu_f((a1 - mean) * rstd * ee_g[lane + 32] + ee_be[lane + 32]);
    }
    __syncthreads();

    // ---- WMMA accumulation: C[16 edges x 64] ----
    int er = e0 + m;
    const float* rp0 = h + (long)dstv[er] * 64 + half * 2;  // W rows   0..63
    const float* rp1 = h + (long)srcv[er] * 64 + half * 2;  // W rows  64..127
    v8f acc0 = {}, acc1 = {}, acc2 = {}, acc3 = {};

#pragma unroll 1
    for (int seg = 0; seg < 2; ++seg) {
        const float* rp = (seg == 0) ? rp0 : rp1;
        const float* wb = W + (long)(seg * 64 + half * 2) * 64 + m;
#pragma unroll 4
        for (int kk = 0; kk < 64; kk += 4) {
            v2f a = *(const v2f*)(rp + kk);
            const float* wp = wb + kk * 64;
            v2f b0 = {wp[0],  wp[64]};
            v2f b1 = {wp[16], wp[80]};
            v2f b2 = {wp[32], wp[96]};
            v2f b3 = {wp[48], wp[112]};
            acc0 = __builtin_amdgcn_wmma_f32_16x16x4_f32(false, a, false, b0, (short)0, acc0, false, false);
            acc1 = __builtin_amdgcn_wmma_f32_16x16x4_f32(false, a, false, b1, (short)0, acc1, false, false);
            acc2 = __builtin_amdgcn_wmma_f32_16x16x4_f32(false, a, false, b2, (short)0, acc2, false, false);
            acc3 = __builtin_amdgcn_wmma_f32_16x16x4_f32(false, a, false, b3, (short)0, acc3, false, false);
        }
    }
    {   // seg 2: edge features from LDS (W rows 128..191)
        const float* wb   = W + (long)(128 + half * 2) * 64 + m;
        const float* lrow = &le[m][half * 2];
#pragma unroll 4
        for (int kk = 0; kk < 64; kk += 4) {
            v2f a = {lrow[kk], lrow[kk + 1]};
            const float* wp = wb + kk * 64;
            v2f b0 = {wp[0],  wp[64]};
            v2f b1 = {wp[16], wp[80]};
            v2f b2 = {wp[32], wp[96]};
            v2f b3 = {wp[48], wp[112]};
            acc0 = __builtin_amdgcn_wmma_f32_16x16x4_f32(false, a, false, b0, (short)0, acc0, false, false);
            acc1 = __builtin_amdgcn_wmma_f32_16x16x4_f32(false, a, false, b1, (short)0, acc1, false, false);
            acc2 = __builtin_amdgcn_wmma_f32_16x16x4_f32(false, a, false, b2, (short)0, acc2, false, false);
            acc3 = __builtin_amdgcn_wmma_f32_16x16x4_f32(false, a, false, b3, (short)0, acc3, false, false);
        }
    }

    // ---- bias -> SiLU -> rowwise LN -> atomic scatter-add into agg[dst] ----
    float bb0 = mb[m], bb1 = mb[16 + m], bb2 = mb[32 + m], bb3 = mb[48 + m];
    float gg0 = mg[m], gg1 = mg[16 + m], gg2 = mg[32 + m], gg3 = mg[48 + m];
    float eb0 = mbe[m], eb1 = mbe[16 + m], eb2 = mbe[32 + m], eb3 = mbe[48 + m];
#pragma unroll
    for (int j = 0; j < 8; ++j) {  // C/D layout: VGPR j holds row j (lanes<16) / j+8
        float v0 = silu_f(acc0[j] + bb0);
        float v1 = silu_f(acc1[j] + bb1);
        float v2 = silu_f(acc2[j] + bb2);
        float v3 = silu_f(acc3[j] + bb3);
        float s = v0 + v1 + v2 + v3, ss = v0 * v0 + v1 * v1 + v2 * v2 + v3 * v3;
#pragma unroll
        for (int xm = 1; xm < 16; xm <<= 1) { s += __shfl_xor(s, xm); ss += __shfl_xor(ss, xm); }
        float mean = s * (1.0f / 64.0f);
        float rstd = rsqrtf(ss * (1.0f / 64.0f) - mean * mean + 1e-5f);
        int dr = dstv[e0 + j + half * 8];
        float* ap = agg + (long)dr * 64 + m;
        atomicAdd(ap,      (v0 - mean) * rstd * gg0 + eb0);
        atomicAdd(ap + 16, (v1 - mean) * rstd * gg1 + eb1);
        atomicAdd(ap + 32, (v2 - mean) * rstd * gg2 + eb2);
        atomicAdd(ap + 48, (v3 - mean) * rstd * gg3 + eb3);
    }
}

// ---------------------------------------------------------------------------
// Update kernel (one wave = 16 nodes):
//   u = LN(SiLU((h+agg) @ W + ub), ug, ube);  h = LN(u + h, ng, nbe)  in-place
// ---------------------------------------------------------------------------
__global__ void __launch_bounds__(128) upd_kernel(
    float* __restrict__ h, const float* __restrict__ agg,
    const float* __restrict__ W,  // [64,64] slice
    const float* __restrict__ ub, const float* __restrict__ ug,
    const float* __restrict__ ube, const float* __restrict__ ng,
    const float* __restrict__ nbe, int ntiles) {
    int tile = blockIdx.x * 4 + (threadIdx.x >> 5);
    if (tile >= ntiles) return;  // wave-uniform; WMMA below runs with full EXEC
    int lane = threadIdx.x & 31;
    int m = lane & 15, half = lane >> 4;
    int n0 = tile * 16;
    const float* ph = h   + (long)(n0 + m) * 64 + half * 2;
    const float* pa = agg + (long)(n0 + m) * 64 + half * 2;
    const float* wb = W + (long)(half * 2) * 64 + m;
    v8f acc0 = {}, acc1 = {}, acc2 = {}, acc3 = {};
#pragma unroll 4
    for (int kk = 0; kk < 64; kk += 4) {
        v2f a  = *(const v2f*)(ph + kk);
        v2f a2 = *(const v2f*)(pa + kk);
        a = a + a2;
        const float* wp = wb + kk * 64;
        v2f b0 = {wp[0],  wp[64]};
        v2f b1 = {wp[16], wp[80]};
        v2f b2 = {wp[32], wp[96]};
        v2f b3 = {wp[48], wp[112]};
        acc0 = __builtin_amdgcn_wmma_f32_16x16x4_f32(false, a, false, b0, (short)0, acc0, false, false);
        acc1 = __builtin_amdgcn_wmma_f32_16x16x4_f32(false, a, false, b1, (short)0, acc1, false, false);
        acc2 = __builtin_amdgcn_wmma_f32_16x16x4_f32(false, a, false, b2, (short)0, acc2, false, false);
        acc3 = __builtin_amdgcn_wmma_f32_16x16x4_f32(false, a, false, b3, (short)0, acc3, false, false);
    }
    float bb0 = ub[m], bb1 = ub[16 + m], bb2 = ub[32 + m], bb3 = ub[48 + m];
    float ug0 = ug[m], ug1 = ug[16 + m], ug2 = ug[32 + m], ug3 = ug[48 + m];
    float ue0 = ube[m], ue1 = ube[16 + m], ue2 = ube[32 + m], ue3 = ube[48 + m];
    float ng0 = ng[m], ng1 = ng[16 + m], ng2 = ng[32 + m], ng3 = ng[48 + m];
    float nb0 = nbe[m], nb1 = nbe[16 + m], nb2 = nbe[32 + m], nb3 = nbe[48 + m];
#pragma unroll
    for (int j = 0; j < 8; ++j) {
        long row = n0 + j + half * 8;
        float* hr = h + row * 64 + m;
        float h0 = hr[0], h1 = hr[16], h2 = hr[32], h3 = hr[48];
        float y0 = silu_f(acc0[j] + bb0);
        float y1 = silu_f(acc1[j] + bb1);
        float y2 = silu_f(acc2[j] + bb2);
        float y3 = silu_f(acc3[j] + bb3);
        float s = y0 + y1 + y2 + y3, ss = y0 * y0 + y1 * y1 + y2 * y2 + y3 * y3;
#pragma unroll
        for (int xm = 1; xm < 16; xm <<= 1) { s += __shfl_xor(s, xm); ss += __shfl_xor(ss, xm); }
        float mean = s * (1.0f / 64.0f);
        float rstd = rsqrtf(ss * (1.0f / 64.0f) - mean * mean + 1e-5f);
        float w0 = (y0 - mean) * rstd * ug0 + ue0 + h0;
        float w1 = (y1 - mean) * rstd * ug1 + ue1 + h1;
        float w2 = (y2 - mean) * rstd * ug2 + ue2 + h2;
        float w3 = (y3 - mean) * rstd * ug3 + ue3 + h3;
        float s2 = w0 + w1 + w2 + w3, ss2 = w0 * w0 + w1 * w1 + w2 * w2 + w3 * w3;
#pragma unroll
        for (int xm = 1; xm < 16; xm <<= 1) { s2 += __shfl_xor(s2, xm); ss2 += __shfl_xor(ss2, xm); }
        float mean2 = s2 * (1.0f / 64.0f);
        float rstd2 = rsqrtf(ss2 * (1.0f / 64.0f) - mean2 * mean2 + 1e-5f);
        hr[0]  = (w0 - mean2) * rstd2 * ng0 + nb0;
        hr[16] = (w1 - mean2) * rstd2 * ng1 + nb1;
        hr[32] = (w2 - mean2) * rstd2 * ng2 + nb2;
        hr[48] = (w3 - mean2) * rstd2 * ng3 + nb3;
    }
}

// ---------------------------------------------------------------------------
// Pool: gsum[batch[n]] += h[n], gcnt[batch[n]] += 1
// ---------------------------------------------------------------------------
__global__ void pool_kernel(const float* __restrict__ h, const int* __restrict__ batch,
                            float* __restrict__ gsum, float* __restrict__ gcnt) {
    long i = (long)blockIdx.x * blockDim.x + threadIdx.x;
    if (i >= (long)NN * 64) return;
    int n = (int)(i >> 6), c = (int)(i & 63);
    int b = batch[n];
    atomicAdd(&gsum[(long)b * 64 + c], h[i]);
    if (c == 0) atomicAdd(&gcnt[b], 1.0f);
}

// ---------------------------------------------------------------------------
// Head: out = SiLU(LN(g @ h1_w + h1_b)) @ h2_w + h2_b    (one 32-lane block/graph)
// ---------------------------------------------------------------------------
__global__ void head_kernel(const float* __restrict__ gsum, const float* __restrict__ gcnt,
                            const float* __restrict__ w1, const float* __restrict__ b1,
                            const float* __restrict__ g1, const float* __restrict__ be1,
                            const float* __restrict__ w2, const float* __restrict__ b2,
                            float* __restrict__ out) {
    int gi = blockIdx.x, lane = threadIdx.x;
    float inv = 1.0f / fmaxf(gcnt[gi], 1.0f);
    const float* gr = gsum + (long)gi * 64;
    float acc = b1[lane];
#pragma unroll 8
    for (int k = 0; k < 64; ++k) acc += gr[k] * inv * w1[k * 32 + lane];
    float s = acc, ss = acc * acc;
#pragma unroll
    for (int xm = 1; xm < 32; xm <<= 1) { s += __shfl_xor(s, xm); ss += __shfl_xor(ss, xm); }
    float mean = s * (1.0f / 32.0f);
    float rstd = rsqrtf(ss * (1.0f / 32.0f) - mean * mean + 1e-5f);
    float z = silu_f((acc - mean) * rstd * g1[lane] + be1[lane]);
    float o = z * w2[lane];
#pragma unroll
    for (int xm = 1; xm < 32; xm <<= 1) o += __shfl_xor(o, xm);
    if (lane == 0) out[gi] = o + b2[0];
}

// ---------------------------------------------------------------------------
extern "C" void kernel_launch(void* const* d_in, const int* in_sizes, int n_in,
                              void* d_out, int out_size, void* d_ws, size_t ws_size,
                              hipStream_t stream) {
    const float* x     = (const float*)d_in[0];
    const float* eattr = (const float*)d_in[1];
    const float* ne_w  = (const float*)d_in[2];
    const float* ne_b  = (const float*)d_in[3];
    const float* ne_g  = (const float*)d_in[4];
    const float* ne_be = (const float*)d_in[5];
    const float* ee_w  = (const float*)d_in[6];
    const float* ee_b  = (const float*)d_in[7];
    const float* ee_g  = (const float*)d_in[8];
    const float* ee_be = (const float*)d_in[9];
    const float* msg_w = (const float*)d_in[10];  // [2,192,64]
    const float* msg_b = (const float*)d_in[11];
    const float* msg_g = (const float*)d_in[12];
    const float* msg_be= (const float*)d_in[13];
    const float* upd_w = (const float*)d_in[14];  // [2,64,64]
    const float* upd_b = (const float*)d_in[15];
    const float* upd_g = (const float*)d_in[16];
    const float* upd_be= (const float*)d_in[17];
    const float* nrm_g = (const float*)d_in[18];
    const float* nrm_be= (const float*)d_in[19];
    const float* h1_w  = (const float*)d_in[20];
    const float* h1_b  = (const float*)d_in[21];
    const float* h1_g  = (const float*)d_in[22];
    const float* h1_be = (const float*)d_in[23];
    const float* h2_w  = (const float*)d_in[24];
    const float* h2_b  = (const float*)d_in[25];
    const int* eidx    = (const int*)d_in[26];    // [2,E]
    const int* batch   = (const int*)d_in[27];
    const int* srcv = eidx;       // edge_index[0]
    const int* dstv = eidx + NE;  // edge_index[1]
    float* out = (float*)d_out;

    // Workspace layout (~25.7 MB of fp32)
    float* h    = (float*)d_ws;
    float* agg  = h + (size_t)NN * 64;
    float* gsum = agg + (size_t)NN * 64;
    float* gcnt = gsum + (size_t)NG * 64;

    // Node embedding (50000 rows, 4 waves/block -> exact grid)
    node_embed_kernel<<<NN / 4, 128, 0, stream>>>(x, ne_w, ne_b, ne_g, ne_be, h, NN);

    for (int l = 0; l < 2; ++l) {
        zero_kernel<<<(NN * 64 + 255) / 256, 256, 0, stream>>>(agg, (long)NN * 64);
        // 50000 edge tiles, 4 waves/block -> 12500 blocks (exact)
        msg_kernel<<<NE / 64, 128, 0, stream>>>(
            h, eattr, ee_w, ee_b, ee_g, ee_be,
            msg_w + (size_t)l * 192 * 64, msg_b + l * 64, msg_g + l * 64, msg_be + l * 64,
            srcv, dstv, agg);
        int ntiles = NN / 16;  // 3125
        upd_kernel<<<(ntiles + 3) / 4, 128, 0, stream>>>(
            h, agg, upd_w + (size_t)l * 64 * 64, upd_b + l * 64,
            upd_g + l * 64, upd_be + l * 64, nrm_g + l * 64, nrm_be + l * 64, ntiles);
    }

    zero_kernel<<<(NG * 65 + 255) / 256, 256, 0, stream>>>(gsum, (long)NG * 65);
    pool_kernel<<<(NN * 64) / 256, 256, 0, stream>>>(h, batch, gsum, gcnt);
    head_kernel<<<NG, 32, 0, stream>>>(gsum, gcnt, h1_w, h1_b, h1_g, h1_be, h2_w, h2_b, out);
}